// KMeans_23313082482801
// MI455X (gfx1250) — compile-verified
//
#include <hip/hip_runtime.h>

typedef __attribute__((ext_vector_type(2))) float v2f;
typedef __attribute__((ext_vector_type(4))) float v4f;
typedef __attribute__((ext_vector_type(8))) float v8f;

#define BQ   131072
#define KC   256
#define DD   16
#define CROW 20   // padded LDS row stride (dwords) for centers: conflict-free, 16B-aligned rows
#define TROW 17   // padded row stride for per-wave score/idx tables

__global__ __launch_bounds__(256) void kmeans_assign(
    const float* __restrict__ X, const float* __restrict__ C,
    float* __restrict__ outIdx, float* __restrict__ outOff)
{
    __shared__ float cLds[KC * CROW];        // centers, row-major, padded
    __shared__ float c2Lds[KC];              // -0.5 * ||c_k||^2
    __shared__ float scoreT[8][16 * TROW];   // per-wave [row][residue] best score
    __shared__ int   idxT[8][16 * TROW];     // per-wave [row][residue] best idx
    __shared__ int   kbest[8][16];           // per-wave final argmin per row

    const int tid  = threadIdx.x;
    const int wave = tid >> 5;
    const int lane = tid & 31;
    const int half = lane >> 4;   // 0: lanes 0-15, 1: lanes 16-31
    const int l16  = lane & 15;

    // ---- Stage all 256x16 centers into LDS; compute -0.5*||c||^2 ----
    {
        const float* crow = C + tid * DD;
        v4f r0 = *(const v4f*)(crow + 0);
        v4f r1 = *(const v4f*)(crow + 4);
        v4f r2 = *(const v4f*)(crow + 8);
        v4f r3 = *(const v4f*)(crow + 12);
        float* dst = cLds + tid * CROW;
        *(v4f*)(dst + 0)  = r0;
        *(v4f*)(dst + 4)  = r1;
        *(v4f*)(dst + 8)  = r2;
        *(v4f*)(dst + 12) = r3;
        float s = 0.0f;
        #pragma unroll
        for (int i = 0; i < 4; ++i)
            s += r0[i]*r0[i] + r1[i]*r1[i] + r2[i]*r2[i] + r3[i]*r3[i];
        c2Lds[tid] = -0.5f * s;
    }
    __syncthreads();

    const int baseRow = blockIdx.x * 128 + wave * 16;
    const int myRow   = baseRow + l16;

    // ---- A fragments: 16x4 f32 per k-step. lane<16: K={4j,4j+1}; lane>=16: K={4j+2,4j+3} ----
    v2f a[4];
    #pragma unroll
    for (int j = 0; j < 4; ++j)
        a[j] = *(const v2f*)(X + myRow * DD + j * 4 + half * 2);

    // ---- Preload this lane's 16 accumulator-init values (-0.5*c2) into registers ----
    float c2reg[16];
    #pragma unroll
    for (int t = 0; t < 16; ++t)
        c2reg[t] = c2Lds[t * 16 + l16];

    float best[8];
    int   bidx[8];
    #pragma unroll
    for (int v = 0; v < 8; ++v) { best[v] = -3.402823466e38f; bidx[v] = 0; }

    // ---- 16 center tiles, fully unrolled: static c2reg indexing (no movrels),
    //      B-fragment DS loads software-pipelined across tiles by the scheduler.
    //      acc init = -0.5*c2 so D = x.c - 0.5*||c||^2 (argmax == argmin d2). ----
    #pragma unroll
    for (int t = 0; t < 16; ++t) {
        const int kcol = t * 16 + l16;           // this lane's center (column N) index
        const float ci = c2reg[t];
        v8f acc = { ci, ci, ci, ci, ci, ci, ci, ci };
        #pragma unroll
        for (int j = 0; j < 4; ++j) {
            // B element (K,N) = centers[N][K]; N = l16, K = 4j + 2*half + {0,1}
            v2f b = *(const v2f*)(cLds + kcol * CROW + j * 4 + half * 2);
            acc = __builtin_amdgcn_wmma_f32_16x16x4_f32(
                      false, a[j], false, b, (short)0, acc, false, false);
        }
        #pragma unroll
        for (int v = 0; v < 8; ++v) {
            if (acc[v] > best[v]) { best[v] = acc[v]; bidx[v] = kcol; }
        }
    }

    // ---- Dump per-lane running best into LDS: row m = v + 8*half, residue n = l16 ----
    #pragma unroll
    for (int v = 0; v < 8; ++v) {
        const int m = v + half * 8;
        scoreT[wave][m * TROW + l16] = best[v];
        idxT[wave][m * TROW + l16]   = bidx[v];
    }
    __syncthreads();

    // ---- Finalize argmin per row (lanes 0-15, one row each); ref tie-break = lowest index ----
    if (lane < 16) {
        float bs = -3.402823466e38f;
        int   bi = 0;
        #pragma unroll
        for (int n = 0; n < 16; ++n) {
            float s = scoreT[wave][lane * TROW + n];
            int   i = idxT[wave][lane * TROW + n];
            if (s > bs || (s == bs && i < bi)) { bs = s; bi = i; }
        }
        outIdx[baseRow + lane] = (float)bi;
        kbest[wave][lane] = bi;
    }
    __syncthreads();

    // ---- Offsets: 2 lanes per row, 8 cols each; fully coalesced float4 stores ----
    {
        const int r  = lane >> 1;
        const int co = (lane & 1) * 8;
        const int k  = kbest[wave][r];
        const float* xr = X + (baseRow + r) * DD + co;
        const float* cr = cLds + k * CROW + co;
        float* orow = outOff + (size_t)(baseRow + r) * DD + co;
        v4f x0 = *(const v4f*)(xr);
        v4f x1 = *(const v4f*)(xr + 4);
        v4f c0 = *(const v4f*)(cr);
        v4f c1 = *(const v4f*)(cr + 4);
        *(v4f*)(orow + 0) = x0 - c0;
        *(v4f*)(orow + 4) = x1 - c1;
    }
}

extern "C" void kernel_launch(void* const* d_in, const int* in_sizes, int n_in,
                              void* d_out, int out_size, void* d_ws, size_t ws_size,
                              hipStream_t stream) {
    const float* X = (const float*)d_in[0];   // trajectory [B, D] f32
    const float* C = (const float*)d_in[1];   // centers    [K, D] f32
    float* outIdx = (float*)d_out;            // [B] indices (as float)
    float* outOff = (float*)d_out + BQ;       // [B, D] offsets
    dim3 grid(BQ / 128), block(256);
    hipLaunchKernelGGL(kmeans_assign, grid, block, 0, stream, X, C, outIdx, outOff);
}